// GCN_scalar_30434138259518
// MI455X (gfx1250) — compile-verified
//
#include <hip/hip_runtime.h>
#include <hip/hip_bf16.h>

typedef __attribute__((ext_vector_type(2))) float v2f;
typedef __attribute__((ext_vector_type(8))) float v8f;

#define N_NODES 100000
#define N_EDGES 1600000
#define FEAT    64

// ---------------------------------------------------------------------------
// Zero a float buffer, float4-vectorized (graph-capture-safe memset).
// ---------------------------------------------------------------------------
__global__ void zero4_kernel(float4* __restrict__ p, long long n4) {
    long long i = (long long)blockIdx.x * blockDim.x + threadIdx.x;
    if (i < n4) p[i] = make_float4(0.f, 0.f, 0.f, 0.f);
}

__global__ void zero1_kernel(float* __restrict__ p, int n) {
    int i = blockIdx.x * blockDim.x + threadIdx.x;
    if (i < n) p[i] = 0.0f;
}

// ---------------------------------------------------------------------------
// Scatter-add aggregation: aggr[dst[e], :] += x[src[e], :]
// One thread per (edge, 4-feature chunk): 16 threads cover an edge's 64
// features. Gather is one global_load_b128 per thread; indices loaded once
// per 4 features. Atomic f32 adds resolve in L2 (aggr = 25.6 MB << 192 MB L2).
// ---------------------------------------------------------------------------
__global__ void scatter_add_kernel(const float* __restrict__ x,
                                   const int* __restrict__ src,
                                   const int* __restrict__ dst,
                                   float* __restrict__ aggr,
                                   int nEdges) {
    long long tid = (long long)blockIdx.x * blockDim.x + threadIdx.x;
    int e = (int)(tid >> 4);       // edge index
    int q = (int)(tid & 15);       // 4-feature chunk index
    if (e >= nEdges) return;
    int s = __ldg(&src[e]);
    int d = __ldg(&dst[e]);
    const float4 v = ((const float4*)x)[(long long)s * (FEAT / 4) + q];
    float* base = aggr + (long long)d * FEAT + q * 4;
    atomicAdd(base + 0, v.x);
    atomicAdd(base + 1, v.y);
    atomicAdd(base + 2, v.z);
    atomicAdd(base + 3, v.w);
}

// ---------------------------------------------------------------------------
// Fused GraphConv dense part with f32 WMMA:
//   out[16x64 tile] = relu( aggr@w_rel + bias + xin@w_root )
// One wave32 per 16-row tile; 16 K-steps of V_WMMA_F32_16X16X4_F32 x 4 column
// tiles x 2 weight matrices = 128 WMMAs/tile.
//
// Weights are staged in LDS once per 256-thread block in a PAIR-INTERLEAVED
// layout matched to the WMMA B fragment:
//   sW2[k2][col] = ( W[2*k2][col], W[2*k2+1][col] )
// so each lane's B operand is a single aligned ds_load_b64 into an even VGPR
// pair (no repacking movs), and all 16 k-steps are immediate-offset DS loads
// off one per-lane base address.
//
// Operand layouts per CDNA5 ISA 7.12.2 (wave32):
//   A (16x4 f32, 2 VGPRs): lanes 0-15 -> M=lane, {K0,K1}; lanes 16-31 -> {K2,K3}
//   B (4x16 f32, 2 VGPRs): lanes 0-15 -> N=lane, {K0,K1}; lanes 16-31 -> {K2,K3}
//   C/D (16x16 f32, 8 VGPRs): VGPR r -> M=r (lanes 0-15) / M=r+8 (lanes 16-31)
// ---------------------------------------------------------------------------
__global__ void __launch_bounds__(256)
gcn_gemm_relu_kernel(const float* __restrict__ aggr,
                     const float* __restrict__ xin,
                     const float* __restrict__ w_rel,
                     const float* __restrict__ b_rel,
                     const float* __restrict__ w_root,
                     float* __restrict__ out,
                     int nTiles) {
    __shared__ v2f sWrel2[(FEAT / 2) * FEAT];    // [k2][col] pairs, 16 KB
    __shared__ v2f sWroot2[(FEAT / 2) * FEAT];   // 16 KB
    __shared__ float sBias[FEAT];

    {   // cooperative stage: 2048 pair-entries per matrix / 256 threads = 8 it
        for (int i = threadIdx.x; i < (FEAT / 2) * FEAT; i += blockDim.x) {
            int k2  = i >> 6;       // i / FEAT
            int col = i & (FEAT - 1);
            v2f pr, po;
            pr.x = w_rel[(long long)(2 * k2) * FEAT + col];
            pr.y = w_rel[(long long)(2 * k2 + 1) * FEAT + col];
            po.x = w_root[(long long)(2 * k2) * FEAT + col];
            po.y = w_root[(long long)(2 * k2 + 1) * FEAT + col];
            sWrel2[i]  = pr;
            sWroot2[i] = po;
        }
        if (threadIdx.x < FEAT) sBias[threadIdx.x] = b_rel[threadIdx.x];
    }
    __syncthreads();

    int wave = (int)(((long long)blockIdx.x * blockDim.x + threadIdx.x) >> 5);
    int lane = threadIdx.x & 31;
    if (wave >= nTiles) return;   // after the only barrier; EXEC all-1s at WMMA

    int row0    = wave * 16;
    int halfSel = lane >> 4;      // 0: lanes 0-15, 1: lanes 16-31
    int lid     = lane & 15;

    const float* aRelBase  = aggr + (long long)(row0 + lid) * FEAT + halfSel * 2;
    const float* aRootBase = xin  + (long long)(row0 + lid) * FEAT + halfSel * 2;

    v8f acc[4];
#pragma unroll
    for (int c = 0; c < 4; ++c) {
        float bv = sBias[c * 16 + lid];   // bias depends only on column N
        acc[c] = (v8f){bv, bv, bv, bv, bv, bv, bv, bv};
    }

    // Per-lane LDS base: k2 = 2*kk + halfSel, entry index = k2*FEAT + col.
    // Base covers (halfSel, col); the kk term is a compile-time immediate.
    const v2f* bRelBase  = sWrel2  + halfSel * FEAT + lid;
    const v2f* bRootBase = sWroot2 + halfSel * FEAT + lid;

#pragma unroll
    for (int kk = 0; kk < 16; ++kk) {
        int kbase = kk * 4;

        v2f aRel  = *(const v2f*)(aRelBase  + kbase);
        v2f aRoot = *(const v2f*)(aRootBase + kbase);

#pragma unroll
        for (int c = 0; c < 4; ++c) {
            v2f bRel  = bRelBase [kk * 2 * FEAT + c * 16];
            acc[c] = __builtin_amdgcn_wmma_f32_16x16x4_f32(
                false, aRel, false, bRel, (short)0, acc[c], false, false);

            v2f bRoot = bRootBase[kk * 2 * FEAT + c * 16];
            acc[c] = __builtin_amdgcn_wmma_f32_16x16x4_f32(
                false, aRoot, false, bRoot, (short)0, acc[c], false, false);
        }
    }

    // ReLU + store: VGPR r holds row M = r + 8*halfSel, column N = c*16 + lid
    int mBase = halfSel * 8;
#pragma unroll
    for (int c = 0; c < 4; ++c) {
#pragma unroll
        for (int r = 0; r < 8; ++r) {
            float v = acc[c][r];
            v = v > 0.0f ? v : 0.0f;
            out[(long long)(row0 + mBase + r) * FEAT + c * 16 + lid] = v;
        }
    }
}

// ---------------------------------------------------------------------------
// Layer 3 (64 -> 1) fused with global mean pool:
//   per-node s = aggr[n,:]@w_rel3 + b_rel3 + h[n,:]@w_root3
//   out[0] = mean(s)   via wave shuffle + LDS + atomicAdd of s/N
// ---------------------------------------------------------------------------
__global__ void final_reduce_kernel(const float* __restrict__ aggr,
                                    const float* __restrict__ h,
                                    const float* __restrict__ w_rel3,
                                    const float* __restrict__ b_rel3,
                                    const float* __restrict__ w_root3,
                                    float* __restrict__ out,
                                    int nNodes) {
    int node = blockIdx.x * blockDim.x + threadIdx.x;
    float val = 0.0f;
    if (node < nNodes) {
        float s = __ldg(&b_rel3[0]);
#pragma unroll 8
        for (int f = 0; f < FEAT; ++f) {
            s += aggr[(long long)node * FEAT + f] * __ldg(&w_rel3[f])
               + h[(long long)node * FEAT + f]    * __ldg(&w_root3[f]);
        }
        val = s;
    }
    // wave32 reduction
    for (int off = 16; off > 0; off >>= 1)
        val += __shfl_down(val, off, 32);

    __shared__ float wsum[8];
    int lane = threadIdx.x & 31;
    int w    = threadIdx.x >> 5;
    if (lane == 0) wsum[w] = val;
    __syncthreads();
    if (threadIdx.x == 0) {
        float bs = 0.0f;
        int nw = (int)(blockDim.x >> 5);
        for (int i = 0; i < nw; ++i) bs += wsum[i];
        atomicAdd(out, bs * (1.0f / (float)N_NODES));
    }
}

// ---------------------------------------------------------------------------
// Launcher
// ---------------------------------------------------------------------------
extern "C" void kernel_launch(void* const* d_in, const int* in_sizes, int n_in,
                              void* d_out, int out_size, void* d_ws, size_t ws_size,
                              hipStream_t stream) {
    const float* x       = (const float*)d_in[0];
    const int*   ei      = (const int*)  d_in[1];   // [2, E]
    const float* w_rel1  = (const float*)d_in[2];
    const float* b_rel1  = (const float*)d_in[3];
    const float* w_root1 = (const float*)d_in[4];
    const float* w_rel2  = (const float*)d_in[5];
    const float* b_rel2  = (const float*)d_in[6];
    const float* w_root2 = (const float*)d_in[7];
    const float* w_rel3  = (const float*)d_in[8];
    const float* b_rel3  = (const float*)d_in[9];
    const float* w_root3 = (const float*)d_in[10];
    float* out = (float*)d_out;

    const int N = N_NODES, E = N_EDGES;
    const int* src = ei;
    const int* dst = ei + E;

    float* aggr = (float*)d_ws;               // N*64 f32 = 25.6 MB
    float* hA   = aggr + (size_t)N * FEAT;    // 25.6 MB
    float* hB   = hA   + (size_t)N * FEAT;    // 25.6 MB

    long long nElem4 = (long long)N * FEAT / 4;
    int zblocks = (int)((nElem4 + 255) / 256);
    int sblocks = (int)(((long long)E * 16 + 255) / 256);
    int nTiles  = N / 16;                     // 6250, exact
    int gblocks = (nTiles + 7) / 8;           // 8 waves (256 thr) per block

    // Layer 1: aggr = scatter(x); hA = relu(aggr@w_rel1 + b1 + x@w_root1)
    zero4_kernel<<<zblocks, 256, 0, stream>>>((float4*)aggr, nElem4);
    scatter_add_kernel<<<sblocks, 256, 0, stream>>>(x, src, dst, aggr, E);
    gcn_gemm_relu_kernel<<<gblocks, 256, 0, stream>>>(aggr, x, w_rel1, b_rel1,
                                                      w_root1, hA, nTiles);
    // Layer 2
    zero4_kernel<<<zblocks, 256, 0, stream>>>((float4*)aggr, nElem4);
    scatter_add_kernel<<<sblocks, 256, 0, stream>>>(hA, src, dst, aggr, E);
    gcn_gemm_relu_kernel<<<gblocks, 256, 0, stream>>>(aggr, hA, w_rel2, b_rel2,
                                                      w_root2, hB, nTiles);
    // Layer 3 + global mean pool
    zero4_kernel<<<zblocks, 256, 0, stream>>>((float4*)aggr, nElem4);
    scatter_add_kernel<<<sblocks, 256, 0, stream>>>(hB, src, dst, aggr, E);
    zero1_kernel<<<1, 32, 0, stream>>>(out, 1);
    final_reduce_kernel<<<(N + 255) / 256, 256, 0, stream>>>(aggr, hB, w_rel3,
                                                             b_rel3, w_root3,
                                                             out, N);
}